// metaCDA_69793218560048
// MI455X (gfx1250) — compile-verified
//
#include <hip/hip_runtime.h>
#include <hip/hip_bf16.h>
#include <math.h>

// ---------------------------------------------------------------------------
// GCN (3-layer symmetric-normalized SpMM over a node subset) + fp32 WMMA MLP
// head for MI455X / gfx1250.
//   Phase 1: degree + edge normalization (loop invariant, computed once)
//   Phase 2: 3x edge-parallel SpMM with f32 atomics (L2-resident, 192MB L2)
//   Phase 3: MLP 128->64->32->21->1 using V_WMMA_F32_16X16X4_F32 (fp32 exact)
// ---------------------------------------------------------------------------

typedef float v2f __attribute__((ext_vector_type(2)));
typedef float v8f __attribute__((ext_vector_type(8)));

#define EPSV 1e-8f

__device__ __forceinline__ v8f wmma4(v2f a, v2f b, v8f c) {
  // D = A(16x4,f32) * B(4x16,f32) + C(16x16,f32)
  return __builtin_amdgcn_wmma_f32_16x16x4_f32(false, a, false, b, (short)0, c,
                                               false, false);
}

__device__ __forceinline__ float leaky(float x) {
  return (x > 0.f) ? x : 0.01f * x;
}

// ---------------- Phase 1: degree + normalized edge values -----------------

__global__ void deg_kernel(const int* __restrict__ row,
                           const float* __restrict__ val,
                           float* __restrict__ deg, int E) {
  int e = blockIdx.x * blockDim.x + threadIdx.x;
  if (e < E) atomicAdd(&deg[row[e]], val[e]);
}

__global__ void nval_kernel(const int* __restrict__ row,
                            const int* __restrict__ col,
                            const float* __restrict__ val,
                            const float* __restrict__ deg,
                            float* __restrict__ nval, int E) {
  int e = blockIdx.x * blockDim.x + threadIdx.x;
  if (e < E) {
    float dr = 1.0f / sqrtf(deg[row[e]] + EPSV);
    float dc = 1.0f / sqrtf(deg[col[e]] + EPSV);
    nval[e] = val[e] * dr * dc;
  }
}

// gather sub0 = features[index], build inverse map inv[index[j]] = j
__global__ void gather_kernel(const float* __restrict__ feat,
                              const int* __restrict__ index,
                              float* __restrict__ sub0, int* __restrict__ inv,
                              int M, int D) {
  long long gid = (long long)blockIdx.x * blockDim.x + threadIdx.x;
  int j = (int)(gid >> 5);
  if (j >= M) return;
  int c = (int)(gid & 31) * 4;            // D=128 -> 32 float4 chunks
  int sr = index[j];
  const float4 s = *(const float4*)(feat + (size_t)sr * D + c);
  *(float4*)(sub0 + (size_t)j * D + c) = s;
  if (c == 0) inv[sr] = j;
}

// ---------------- Phase 2: edge-parallel SpMM (scatter-add) ----------------

__global__ void spmm_kernel(const int* __restrict__ row,
                            const int* __restrict__ col,
                            const float* __restrict__ nval,
                            const float* __restrict__ src,
                            float* __restrict__ dst, int E, int D) {
  long long gid = (long long)blockIdx.x * blockDim.x + threadIdx.x;
  int e = (int)(gid >> 5);
  if (e >= E) return;
  int c = (int)(gid & 31) * 4;
  const float nv = nval[e];
  const float4 s = *(const float4*)(src + (size_t)col[e] * D + c);
  float* d = dst + (size_t)row[e] * D + c;
  atomicAdd(d + 0, s.x * nv);
  atomicAdd(d + 1, s.y * nv);
  atomicAdd(d + 2, s.z * nv);
  atomicAdd(d + 3, s.w * nv);
}

// ---------------- Phase 3: fp32 WMMA MLP head ------------------------------
// One wave per 16x16 output tile. A layout: lane L<16 -> row L, K={0,1};
// lanes 16-31 -> same rows, K={2,3}. B layout: col = lane&15, K-pair =
// (lane>>4)*2. C/D: M = v + 8*(lane>>4), N = lane&15.

__global__ __launch_bounds__(128) void mlp1_kernel(
    const float* __restrict__ feat, const float* __restrict__ sub3,
    const int* __restrict__ inv, const float* __restrict__ W1,
    const float* __restrict__ b1, float* __restrict__ h1, int N, int D,
    int D2) {
  const int lane = threadIdx.x & 31;
  const int wave = threadIdx.x >> 5;
  const int row0 = blockIdx.x * 16;
  const int n0 = wave * 16;
  if (n0 >= D2) return;                    // uniform per wave
  const int mrow = lane & 15;
  const int koff = (lane >> 4) * 2;        // 0 or 2
  int r = row0 + mrow;
  if (r >= N) r = N - 1;                   // clamp keeps EXEC full
  const int j = inv[r];
  const float* src = (j >= 0) ? (sub3 + (size_t)j * D) : (feat + (size_t)r * D);
  const int ncol = n0 + mrow;
  v8f acc = {};
  for (int kb = 0; kb < D; kb += 4) {
    v2f a, b;
    a.x = src[kb + koff + 0];
    a.y = src[kb + koff + 1];
    b.x = W1[(size_t)(kb + koff + 0) * D2 + ncol];
    b.y = W1[(size_t)(kb + koff + 1) * D2 + ncol];
    acc = wmma4(a, b, acc);
  }
  const float bias = b1[ncol];
  const int mbase = (lane >> 4) * 8;
#pragma unroll
  for (int v = 0; v < 8; ++v) {
    int rr = row0 + mbase + v;
    if (rr < N) h1[(size_t)rr * D2 + ncol] = leaky(acc[v] + bias);
  }
}

__global__ __launch_bounds__(64) void mlp2_kernel(
    const float* __restrict__ h1, const float* __restrict__ W2,
    const float* __restrict__ b2, float* __restrict__ h2, int N, int D2,
    int D3) {
  const int lane = threadIdx.x & 31;
  const int wave = threadIdx.x >> 5;
  const int row0 = blockIdx.x * 16;
  const int n0 = wave * 16;
  if (n0 >= D3) return;
  const int mrow = lane & 15;
  const int koff = (lane >> 4) * 2;
  int r = row0 + mrow;
  if (r >= N) r = N - 1;
  const float* src = h1 + (size_t)r * D2;
  const int ncol = n0 + mrow;
  v8f acc = {};
  for (int kb = 0; kb < D2; kb += 4) {
    v2f a, b;
    a.x = src[kb + koff + 0];
    a.y = src[kb + koff + 1];
    b.x = W2[(size_t)(kb + koff + 0) * D3 + ncol];
    b.y = W2[(size_t)(kb + koff + 1) * D3 + ncol];
    acc = wmma4(a, b, acc);
  }
  const float bias = b2[ncol];
  const int mbase = (lane >> 4) * 8;
#pragma unroll
  for (int v = 0; v < 8; ++v) {
    int rr = row0 + mbase + v;
    if (rr < N) h2[(size_t)rr * D3 + ncol] = leaky(acc[v] + bias);
  }
}

// D4=21 padded to HP=32 output columns; pad columns get zeros.
__global__ __launch_bounds__(64) void mlp3_kernel(
    const float* __restrict__ h2, const float* __restrict__ W3,
    const float* __restrict__ b3, float* __restrict__ h3, int N, int D3,
    int D4, int HP) {
  const int lane = threadIdx.x & 31;
  const int wave = threadIdx.x >> 5;
  const int row0 = blockIdx.x * 16;
  const int n0 = wave * 16;
  if (n0 >= HP) return;
  const int mrow = lane & 15;
  const int koff = (lane >> 4) * 2;
  int r = row0 + mrow;
  if (r >= N) r = N - 1;
  const float* src = h2 + (size_t)r * D3;
  const int ncol = n0 + mrow;
  const bool cok = (ncol < D4);
  v8f acc = {};
  for (int kb = 0; kb < D3; kb += 4) {
    v2f a, b;
    a.x = src[kb + koff + 0];
    a.y = src[kb + koff + 1];
    b.x = cok ? W3[(size_t)(kb + koff + 0) * D4 + ncol] : 0.f;
    b.y = cok ? W3[(size_t)(kb + koff + 1) * D4 + ncol] : 0.f;
    acc = wmma4(a, b, acc);
  }
  const float bias = cok ? b3[ncol] : 0.f;
  const int mbase = (lane >> 4) * 8;
#pragma unroll
  for (int v = 0; v < 8; ++v) {
    int rr = row0 + mbase + v;
    if (rr < N) h3[(size_t)rr * HP + ncol] = leaky(acc[v] + bias);
  }
}

__global__ void mlp4_kernel(const float* __restrict__ h3,
                            const float* __restrict__ W4,
                            float* __restrict__ out, int N, int D4, int HP) {
  int i = blockIdx.x * blockDim.x + threadIdx.x;
  if (i >= N) return;
  const float* src = h3 + (size_t)i * HP;
  float acc = 0.f;
  for (int k = 0; k < D4; ++k) acc += src[k] * W4[k];
  out[i] = 1.0f / (1.0f + expf(-acc));
}

// ---------------------------------------------------------------------------

extern "C" void kernel_launch(void* const* d_in, const int* in_sizes, int n_in,
                              void* d_out, int out_size, void* d_ws,
                              size_t ws_size, hipStream_t stream) {
  const float* feat  = (const float*)d_in[0];
  const int*   index = (const int*)d_in[1];
  const int*   erow  = (const int*)d_in[2];
  const int*   ecol  = (const int*)d_in[3];
  const float* evalp = (const float*)d_in[4];
  const float* W1 = (const float*)d_in[5];
  const float* b1 = (const float*)d_in[6];
  const float* W2 = (const float*)d_in[7];
  const float* b2 = (const float*)d_in[8];
  const float* W3 = (const float*)d_in[9];
  const float* b3 = (const float*)d_in[10];
  const float* W4 = (const float*)d_in[11];
  float* out = (float*)d_out;

  const int D  = 128;                       // embedding size (reference)
  const int N  = in_sizes[0] / D;
  const int M  = in_sizes[1];
  const int E  = in_sizes[2];
  const int D2 = in_sizes[6];               // 64
  const int D3 = in_sizes[8];               // 32
  const int D4 = in_sizes[10];              // 21
  const int HP = ((D4 + 15) / 16) * 16;     // 32 (padded)

  // ---- workspace carve-up (256B aligned) ----
  char* ws = (char*)d_ws;
  size_t off = 0;
  auto carve = [&](size_t bytes) {
    size_t o = off;
    off = (off + bytes + 255) & ~(size_t)255;
    return o;
  };
  size_t subf = (size_t)M * D;
  size_t bufA_f = subf;                              // also h1 (N*D2), h3 (N*HP)
  if ((size_t)N * D2 > bufA_f) bufA_f = (size_t)N * D2;
  if ((size_t)N * HP > bufA_f) bufA_f = (size_t)N * HP;
  size_t bufB_f = subf;                              // also h2 (N*D3)
  if ((size_t)N * D3 > bufB_f) bufB_f = (size_t)N * D3;

  float* deg  = (float*)(ws + carve((size_t)M * 4));
  float* nval = (float*)(ws + carve((size_t)E * 4));
  int*   inv  = (int*)  (ws + carve((size_t)N * 4));
  float* bufA = (float*)(ws + carve(bufA_f * 4));
  float* bufB = (float*)(ws + carve(bufB_f * 4));
  (void)ws_size; (void)n_in; (void)out_size;

  // ---- Phase 1: degrees + normalized edge weights (loop invariant) ----
  hipMemsetAsync(deg, 0, (size_t)M * 4, stream);
  hipMemsetAsync(inv, 0xFF, (size_t)N * 4, stream);   // inv = -1
  int eb = (E + 255) / 256;
  deg_kernel<<<eb, 256, 0, stream>>>(erow, evalp, deg, E);
  nval_kernel<<<eb, 256, 0, stream>>>(erow, ecol, evalp, deg, nval, E);

  long long gth = (long long)M * 32;                  // 32 lanes * float4 = 128
  gather_kernel<<<(int)((gth + 255) / 256), 256, 0, stream>>>(feat, index, bufA,
                                                              inv, M, D);

  // ---- Phase 2: 3 SpMM layers, ping-pong bufA <-> bufB ----
  long long sth = (long long)E * 32;
  int sblocks = (int)((sth + 255) / 256);
  float* src = bufA;
  float* dst = bufB;
  for (int l = 0; l < 3; ++l) {
    hipMemsetAsync(dst, 0, subf * 4, stream);
    spmm_kernel<<<sblocks, 256, 0, stream>>>(erow, ecol, nval, src, dst, E, D);
    float* t = src; src = dst; dst = t;
  }
  const float* sub3 = src;                            // after 3 swaps: bufB
  float* h1 = dst;                                    // bufA (dead sub2)
  float* h2 = (h1 == bufA) ? bufB : bufA;
  float* h3 = h1;                                     // bufA again (h1 dead)

  // ---- Phase 3: WMMA fp32 MLP head ----
  int rt = (N + 15) / 16;
  mlp1_kernel<<<rt, 32 * ((D2 + 15) / 16), 0, stream>>>(feat, sub3, inv, W1, b1,
                                                        h1, N, D, D2);
  mlp2_kernel<<<rt, 32 * ((D3 + 15) / 16), 0, stream>>>(h1, W2, b2, h2, N, D2,
                                                        D3);
  mlp3_kernel<<<rt, 32 * (HP / 16), 0, stream>>>(h2, W3, b3, h3, N, D3, D4, HP);
  mlp4_kernel<<<(N + 255) / 256, 256, 0, stream>>>(h3, W4, out, N, D4, HP);
}